// RegionProposalNetwork_89197880803948
// MI455X (gfx1250) — compile-verified
//
#include <hip/hip_runtime.h>
#include <math.h>

// ---------------- problem constants (from reference) ----------------
#define RPN_N      8
#define RPN_A      15
#define RPN_H      100
#define RPN_W      100
#define RPN_HW     (RPN_H * RPN_W)          // 10000
#define RPN_HWA    (RPN_HW * RPN_A)         // 150000
#define PRE_NMS    2000
#define K_PAD      2048
#define POST_NMS   1000
#define NMS_THRESH 0.7f
#define MIN_SIZE   1e-3f
#define IMG_SIZE   1600.0f
#define BBOX_CLIP  4.135166556742356f       // log(1000/16)
#define STRIDE_PX  16.0f

// ---------------- TDM support detection ----------------
#if defined(__has_builtin)
#if __has_builtin(__builtin_amdgcn_tensor_load_to_lds) && __has_builtin(__builtin_amdgcn_s_wait_tensorcnt)
#define HAVE_TDM 1
#endif
#endif

#ifdef HAVE_TDM
typedef unsigned int u32x4 __attribute__((ext_vector_type(4)));
typedef int          i32x8 __attribute__((ext_vector_type(8)));
typedef int          i32x4 __attribute__((ext_vector_type(4)));

// 1-D contiguous dword copy: global -> LDS via Tensor Data Mover.
// ndwords must be <= 65535 (tile_dim0 is 16-bit).
// 6-arg builtin form (clang-23 / therock-10.0 headers):
//   (u32x4 g0, i32x8 g1, i32x4 g2, i32x4 g3, i32x8 extra, i32 cpol)
__device__ __forceinline__ void tdm_load_1d(unsigned lds_byte_addr,
                                            const void* gptr,
                                            unsigned ndwords) {
    unsigned long long ga = (unsigned long long)(uintptr_t)gptr;
    u32x4 g0;
    g0.x = 1u;                                    // count=1, is_restore=0, gather off
    g0.y = lds_byte_addr;                         // lds_addr [63:32]
    g0.z = (unsigned)(ga & 0xFFFFFFFFull);        // global_addr[31:0]
    g0.w = (unsigned)((ga >> 32) & 0x1FFFFFFull)  // global_addr[56:32]
         | (2u << 30);                            // type = 2 ("image")
    i32x8 g1;
    g1[0] = (int)(2u << 16);                      // data_size = 2 -> 4 bytes
    g1[1] = (int)((ndwords & 0xFFFFu) << 16);     // tensor_dim0[15:0] @ [63:48]
    g1[2] = (int)(((ndwords >> 16) & 0xFFFFu)     // tensor_dim0[31:16]
         | (1u << 16));                           // tensor_dim1 = 1
    g1[3] = (int)((ndwords & 0xFFFFu) << 16);     // tile_dim0 @ [127:112]
    g1[4] = 1;                                    // tile_dim1 = 1, tile_dim2 = 0
    g1[5] = (int)ndwords;                         // tensor_dim0_stride[31:0]
    g1[6] = 0;                                    // stride0[47:32], stride1[15:0]
    g1[7] = 0;
    i32x4 z4; z4[0] = 0; z4[1] = 0; z4[2] = 0; z4[3] = 0;
    i32x8 z8; z8[0] = 0; z8[1] = 0; z8[2] = 0; z8[3] = 0;
              z8[4] = 0; z8[5] = 0; z8[6] = 0; z8[7] = 0;
    __builtin_amdgcn_tensor_load_to_lds(g0, g1, z4, z4, z8, 0);
}
#endif

// ---------------- anchor helper (matches jnp.round / round-to-even) -------
__device__ __forceinline__ void anchor_wh(int a, float& wa, float& ha) {
    int r = a / 5;              // ratio index: 0.5, 1.0, 2.0
    int s = a % 5;              // scale index: 32..512
    float scale = 32.0f * (float)(1 << s);
    float hr = (r == 0) ? 0.70710678118654752f : ((r == 1) ? 1.0f : 1.41421356237309515f);
    float wr = 1.0f / hr;
    wa = 2.0f * rintf(0.5f * wr * scale);   // x2 - x1 of cell anchor
    ha = 2.0f * rintf(0.5f * hr * scale);
}

// =====================================================================
// Kernel 1: sigmoid(objectness), permute [N,A,H,W] -> [N, HW*A]
// =====================================================================
__global__ __launch_bounds__(256)
void rpn_scores_kernel(const float* __restrict__ obj, float* __restrict__ scoresAll) {
    int tid = blockIdx.x * blockDim.x + threadIdx.x;
    if (tid >= RPN_N * RPN_HWA) return;
    int n  = tid / RPN_HWA;
    int r  = tid - n * RPN_HWA;      // layout a*HW + hw (coalesced read)
    int a  = r / RPN_HW;
    int hw = r - a * RPN_HW;
    float x = obj[tid];
    float s = 1.0f / (1.0f + expf(-x));
    scoresAll[(size_t)n * RPN_HWA + (size_t)hw * RPN_A + a] = s;
}

// =====================================================================
// Kernel 2: per-image exact top-2000 via 4-pass radix-256 select.
// One 1024-thread block per image. Scores are sigmoid outputs (>0), so
// the raw float bit pattern is order-preserving as an unsigned int.
// =====================================================================
__global__ __launch_bounds__(1024)
void rpn_topk_kernel(const float* __restrict__ scoresAll,
                     float* __restrict__ selScore,
                     unsigned* __restrict__ selIdx) {
    int n = blockIdx.x;
    int tid = threadIdx.x;
    const float* sc = scoresAll + (size_t)n * RPN_HWA;

    __shared__ unsigned hist[256];
    __shared__ unsigned shPrefix, shRem;

    if (tid == 0) { shPrefix = 0u; shRem = PRE_NMS; }
    __syncthreads();

    for (int round = 0; round < 4; ++round) {
        int shift = 24 - 8 * round;
        if (tid < 256) hist[tid] = 0u;
        __syncthreads();
        unsigned pfx = shPrefix;
        for (int i = tid; i < RPN_HWA; i += 1024) {
            unsigned key = __float_as_uint(sc[i]);
            if (round == 0 || (key >> (shift + 8)) == pfx)
                atomicAdd(&hist[(key >> shift) & 255u], 1u);
        }
        __syncthreads();
        if (tid == 0) {
            unsigned c = 0; int d = 0;
            for (d = 255; d >= 0; --d) {
                if (c + hist[d] >= shRem) break;
                c += hist[d];
            }
            if (d < 0) d = 0;                 // defensive; cannot happen
            shRem -= c;
            shPrefix = (pfx << 8) | (unsigned)d;
        }
        __syncthreads();
    }

    __shared__ unsigned cntG, cntE;
    if (tid == 0) { cntG = 0u; cntE = 0u; }
    __syncthreads();
    unsigned T   = shPrefix;            // exact threshold key
    unsigned rem = shRem;               // # of ties (== T) to accept
    unsigned G   = PRE_NMS - rem;       // # strictly greater than T

    float*    os = selScore + (size_t)n * K_PAD;
    unsigned* oi = selIdx   + (size_t)n * K_PAD;

    for (int i = tid; i < RPN_HWA; i += 1024) {
        float v = sc[i];
        unsigned key = __float_as_uint(v);
        if (key > T) {
            unsigned p = atomicAdd(&cntG, 1u);
            os[p] = v; oi[p] = (unsigned)i;
        } else if (key == T) {
            unsigned p = atomicAdd(&cntE, 1u);
            if (p < rem) { os[G + p] = v; oi[G + p] = (unsigned)i; }
        }
    }
    // pad 2000..2047 with sentinels
    for (int i = PRE_NMS + tid; i < K_PAD; i += 1024) {
        os[i] = -1e30f; oi[i] = 0u;
    }
}

// =====================================================================
// Kernel 3: decode + clip boxes for the selected anchors only.
// =====================================================================
__global__ __launch_bounds__(256)
void rpn_decode_sel_kernel(const float* __restrict__ deltas,
                           const unsigned* __restrict__ selIdx,
                           float4* __restrict__ selBox) {
    int tid = blockIdx.x * blockDim.x + threadIdx.x;
    if (tid >= RPN_N * K_PAD) return;
    int n = tid >> 11;
    int s = tid & (K_PAD - 1);
    float4 box = make_float4(0.f, 0.f, 0.f, 0.f);
    if (s < PRE_NMS) {
        unsigned f = selIdx[(size_t)n * K_PAD + s];
        int a  = (int)(f % RPN_A);
        int hw = (int)(f / RPN_A);
        int h  = hw / RPN_W;
        int w  = hw - h * RPN_W;
        float wa, ha; anchor_wh(a, wa, ha);
        float cxa = (float)w * STRIDE_PX;
        float cya = (float)h * STRIDE_PX;
        size_t base = (((size_t)n * RPN_A + a) * 4) * RPN_HW + hw;
        float dx = deltas[base];
        float dy = deltas[base + RPN_HW];
        float dw = fminf(deltas[base + 2 * (size_t)RPN_HW], BBOX_CLIP);
        float dh = fminf(deltas[base + 3 * (size_t)RPN_HW], BBOX_CLIP);
        float cx = dx * wa + cxa;
        float cy = dy * ha + cya;
        float bw = expf(dw) * wa;
        float bh = expf(dh) * ha;
        box.x = fminf(fmaxf(cx - 0.5f * bw, 0.f), IMG_SIZE);
        box.y = fminf(fmaxf(cy - 0.5f * bh, 0.f), IMG_SIZE);
        box.z = fminf(fmaxf(cx + 0.5f * bw, 0.f), IMG_SIZE);
        box.w = fminf(fmaxf(cy + 0.5f * bh, 0.f), IMG_SIZE);
    }
    selBox[(size_t)n * K_PAD + s] = box;
}

// =====================================================================
// Kernel 4: per image — TDM stage to LDS, bitonic sort 2048 descending,
// greedy NMS (serial over i, parallel over j; matches the reference
// lax.scan exactly), compaction, [1000,5] output with zero padding.
// One 1024-thread (32-wave) block per image; ~62 KB LDS resident state.
// =====================================================================
__global__ __launch_bounds__(1024)
void rpn_sort_nms_kernel(const float* __restrict__ selScore,
                         const float4* __restrict__ selBox,
                         float* __restrict__ out) {
    int n = blockIdx.x;
    int tid = threadIdx.x;

    __shared__ float          key[K_PAD];        //  8 KB  sorted scores
    __shared__ float4         boxA[K_PAD];       // 32 KB  boxes (pre-sort order)
    __shared__ unsigned short idx[K_PAD];        //  4 KB  sort payload
    __shared__ float          area[K_PAD];       //  8 KB  area in sorted order
    __shared__ unsigned char  keep[K_PAD];       //  2 KB
    __shared__ unsigned       pfx[K_PAD];        //  8 KB  output positions

    // ---- stage scores + boxes into LDS ----
#ifdef HAVE_TDM
    if (tid < 32) {   // one wave issues the DMA; TENSORcnt is per-wave
        tdm_load_1d((unsigned)(uintptr_t)(void*)&key[0],
                    selScore + (size_t)n * K_PAD, K_PAD);
        tdm_load_1d((unsigned)(uintptr_t)(void*)&boxA[0],
                    selBox + (size_t)n * K_PAD, K_PAD * 4);
        __builtin_amdgcn_s_wait_tensorcnt(0);
    }
#else
    for (int i = tid; i < K_PAD; i += 1024) {
        key[i]  = selScore[(size_t)n * K_PAD + i];
        boxA[i] = selBox[(size_t)n * K_PAD + i];
    }
#endif
    for (int i = tid; i < K_PAD; i += 1024) idx[i] = (unsigned short)i;
    // zero output slab for this image while DMA is in flight elsewhere
    float* o = out + (size_t)n * POST_NMS * 5;
    for (int i = tid; i < POST_NMS * 5; i += 1024) o[i] = 0.0f;
    __threadfence_block();
    __syncthreads();

    // ---- bitonic sort, descending by key, payload = idx ----
    for (int k = 2; k <= K_PAD; k <<= 1) {
        for (int j = k >> 1; j > 0; j >>= 1) {
            int i0 = 2 * tid - (tid & (j - 1));
            int i1 = i0 + j;
            bool up = ((i0 & k) == 0);
            float ka = key[i0], kb = key[i1];
            bool doswap = up ? (ka < kb) : (ka > kb);
            if (doswap) {
                key[i0] = kb; key[i1] = ka;
                unsigned short t = idx[i0]; idx[i0] = idx[i1]; idx[i1] = t;
            }
            __syncthreads();
        }
    }

    // ---- area + validity in sorted order ----
    for (int i = tid; i < K_PAD; i += 1024) {
        float4 b = boxA[idx[i]];
        float bw = b.z - b.x, bh = b.w - b.y;
        area[i] = bw * bh;
        keep[i] = (key[i] > -0.5f) && (bw >= MIN_SIZE) && (bh >= MIN_SIZE);
    }
    __syncthreads();

    // per-thread register cache of its two j-slots
    int j0 = tid, j1 = tid + 1024;
    float4 b0 = boxA[idx[j0]], b1 = boxA[idx[j1]];
    float  a0 = area[j0],      a1 = area[j1];

    // ---- greedy NMS: serial over i, parallel over j > i ----
    for (int i = 0; i < K_PAD - 1; ++i) {
        if (keep[i]) {
            float4 bi = boxA[idx[i]];
            float  ai = area[i];
            if (j0 > i && keep[j0]) {
                float lx = fmaxf(bi.x, b0.x), ly = fmaxf(bi.y, b0.y);
                float rx = fminf(bi.z, b0.z), ry = fminf(bi.w, b0.w);
                float inter = fmaxf(rx - lx, 0.f) * fmaxf(ry - ly, 0.f);
                float iou = inter / (ai + a0 - inter + 1e-6f);
                if (iou > NMS_THRESH) keep[j0] = 0;
            }
            if (j1 > i && keep[j1]) {
                float lx = fmaxf(bi.x, b1.x), ly = fmaxf(bi.y, b1.y);
                float rx = fminf(bi.z, b1.z), ry = fminf(bi.w, b1.w);
                float inter = fmaxf(rx - lx, 0.f) * fmaxf(ry - ly, 0.f);
                float iou = inter / (ai + a1 - inter + 1e-6f);
                if (iou > NMS_THRESH) keep[j1] = 0;
            }
        }
        __syncthreads();
    }

    // ---- exclusive prefix sum of keep flags (serial: 2048 LDS adds) ----
    if (tid == 0) {
        unsigned c = 0;
        for (int i = 0; i < K_PAD; ++i) { pfx[i] = c; c += keep[i]; }
    }
    __syncthreads();

    // ---- emit kept boxes in sorted order, capped at POST_NMS ----
    if (keep[j0] && pfx[j0] < POST_NMS) {
        unsigned p = pfx[j0];
        o[p * 5 + 0] = b0.x; o[p * 5 + 1] = b0.y;
        o[p * 5 + 2] = b0.z; o[p * 5 + 3] = b0.w;
        o[p * 5 + 4] = key[j0];
    }
    if (keep[j1] && pfx[j1] < POST_NMS) {
        unsigned p = pfx[j1];
        o[p * 5 + 0] = b1.x; o[p * 5 + 1] = b1.y;
        o[p * 5 + 2] = b1.z; o[p * 5 + 3] = b1.w;
        o[p * 5 + 4] = key[j1];
    }
}

// =====================================================================
// host-side launcher
// =====================================================================
extern "C" void kernel_launch(void* const* d_in, const int* in_sizes, int n_in,
                              void* d_out, int out_size, void* d_ws, size_t ws_size,
                              hipStream_t stream) {
    (void)in_sizes; (void)n_in; (void)out_size; (void)ws_size;
    const float* obj    = (const float*)d_in[0];   // [8,15,100,100]
    const float* deltas = (const float*)d_in[1];   // [8,60,100,100]
    float* out = (float*)d_out;                    // [8,1000,5]

    // workspace layout (floats), all 16B aligned:
    float*    scoresAll = (float*)d_ws;                            // N*HWA     (4.8 MB)
    float*    selScore  = scoresAll + (size_t)RPN_N * RPN_HWA;     // N*2048
    unsigned* selIdx    = (unsigned*)(selScore + (size_t)RPN_N * K_PAD);
    float4*   selBox    = (float4*)(selIdx + (size_t)RPN_N * K_PAD);

    {
        int total = RPN_N * RPN_HWA;
        rpn_scores_kernel<<<(total + 255) / 256, 256, 0, stream>>>(obj, scoresAll);
    }
    rpn_topk_kernel<<<RPN_N, 1024, 0, stream>>>(scoresAll, selScore, selIdx);
    {
        int total = RPN_N * K_PAD;
        rpn_decode_sel_kernel<<<(total + 255) / 256, 256, 0, stream>>>(deltas, selIdx, selBox);
    }
    rpn_sort_nms_kernel<<<RPN_N, 1024, 0, stream>>>(selScore, selBox, out);
}